// AdaptiveSampler_31791347925003
// MI455X (gfx1250) — compile-verified
//
#include <hip/hip_runtime.h>
#include <math.h>

#define B_SZ   4096
#define NCAND  512
#define DIM    128
#define KSAMP  32
#define GAMMA  0.1f

typedef float v2f __attribute__((ext_vector_type(2)));
typedef float v8f __attribute__((ext_vector_type(8)));

// ---------------------------------------------------------------------------
// Kernel 0: fold both linear layers into one matrix:
//   M[f][d] = sum_e Wq[e][f] * Wk[e][d]   (= Wq^T @ Wk)
//   c[d]    = sum_e bq[e]    * Wk[e][d]
// Tiny (4 MFLOP) - plain VALU.
// ---------------------------------------------------------------------------
__global__ __launch_bounds__(128) void as_precompute_kernel(
    const float* __restrict__ Wq, const float* __restrict__ bq,
    const float* __restrict__ Wk, float* __restrict__ Mmat,
    float* __restrict__ cvec)
{
    const int f = blockIdx.x;
    const int d = threadIdx.x;
    float acc = 0.f;
    for (int e = 0; e < DIM; ++e)
        acc = fmaf(Wq[e * DIM + f], Wk[e * DIM + d], acc);
    Mmat[f * DIM + d] = acc;
    if (f == 0) {
        float a2 = 0.f;
        for (int e = 0; e < DIM; ++e)
            a2 = fmaf(bq[e], Wk[e * DIM + d], a2);
        cvec[d] = a2;
    }
}

// ---------------------------------------------------------------------------
// Kernel 1: V = T @ M + c   ([4096x128] @ [128x128])  via V_WMMA_F32_16X16X4_F32
// One wave per 16x16 output tile, K=128 in 32 steps of 4.
// VGPR layouts per CDNA5 ISA 7.12.2:
//   A (16x4 f32):  lanes 0-15 -> M=lane, K={0,1}; lanes 16-31 -> M=lane-16, K={2,3}
//   B (4x16 f32):  lanes 0-15 -> N=lane, K={0,1}; lanes 16-31 -> N=lane-16, K={2,3}
//   C/D (16x16):   VGPR i -> row M = i + 8*(lane/16), col N = lane%16
// ---------------------------------------------------------------------------
__global__ __launch_bounds__(32) void as_project_kernel(
    const float* __restrict__ T, const float* __restrict__ Mmat,
    const float* __restrict__ cvec, float* __restrict__ V)
{
    const int lane = threadIdx.x;
    const int half = lane >> 4;     // 0: K pair {0,1}, 1: K pair {2,3}
    const int lm   = lane & 15;
    const int brow = blockIdx.x * 16;
    const int dcol = blockIdx.y * 16;

    v8f acc = {};
    const float* arow = T + (size_t)(brow + lm) * DIM;

    for (int kk = 0; kk < DIM; kk += 4) {
        const int ka = kk + 2 * half;
        v2f a, b;
        a.x = arow[ka];
        a.y = arow[ka + 1];
        b.x = Mmat[(size_t)ka * DIM + dcol + lm];
        b.y = Mmat[(size_t)(ka + 1) * DIM + dcol + lm];
        acc = __builtin_amdgcn_wmma_f32_16x16x4_f32(
            /*neg_a=*/false, a, /*neg_b=*/false, b,
            /*c_mod=*/(short)0, acc, /*reuse_a=*/false, /*reuse_b=*/false);
    }

    const float cadd = cvec[dcol + lm];
#pragma unroll
    for (int i = 0; i < 8; ++i) {
        const int m = i + 8 * half;
        V[(size_t)(brow + m) * DIM + dcol + lm] = acc[i] + cadd;
    }
}

// ---------------------------------------------------------------------------
// Kernel 2: the bandwidth kernel. One block (512 thr = 16 waves) per batch b.
//   scores[n] = (v[b] . cand[b,n]) / sqrt(128)    (streams 256 KB per block)
//   softmax -> mix with uniform -> log(p) + Gumbel(u) -> iterative top-32.
// Lane-per-k layout: lane l holds v[4l..4l+3]; each candidate row is one
// fully-coalesced 512B global_load_b128 across the wave + butterfly reduce.
// ---------------------------------------------------------------------------
__global__ __launch_bounds__(512) void as_sampler_kernel(
    const float* __restrict__ cand, const float* __restrict__ V,
    const float* __restrict__ u, int* __restrict__ out)
{
    const int b    = blockIdx.x;
    const int t    = threadIdx.x;
    const int wave = t >> 5;
    const int lane = t & 31;

    __shared__ float s_sc[NCAND];
    __shared__ float s_red[16];
    __shared__ int   s_ridx[16];
    __shared__ float s_b;
    __shared__ int   s_widx;

    // per-lane slice of v[b]
    const float4 v4 = *(const float4*)(V + (size_t)b * DIM + lane * 4);
    const float* cb = cand + (size_t)b * NCAND * DIM;
    const float scale = 0.08838834764831845f;   // 1/sqrt(128)

    // ---- batched dot products: wave w handles candidates [32w, 32w+32) ----
    for (int j = 0; j < 32; ++j) {
        const int n = wave * 32 + j;
        const float* row = cb + (size_t)n * DIM + lane * 4;
        if (j + 4 < 32) __builtin_prefetch(row + 4 * DIM, 0, 0);
        const float4 c4 = *(const float4*)row;
        float p = c4.x * v4.x + c4.y * v4.y + c4.z * v4.z + c4.w * v4.w;
#pragma unroll
        for (int off = 16; off > 0; off >>= 1) p += __shfl_xor(p, off, 32);
        if (lane == 0) s_sc[n] = p * scale;
    }
    __syncthreads();

    // ---- softmax over 512 (thread t owns candidate t) ----
    const float sc = s_sc[t];
    float m = sc;
#pragma unroll
    for (int off = 16; off > 0; off >>= 1) m = fmaxf(m, __shfl_xor(m, off, 32));
    if (lane == 0) s_red[wave] = m;
    __syncthreads();
    if (t < 16) {
        float v2 = s_red[t];
#pragma unroll
        for (int off = 8; off > 0; off >>= 1) v2 = fmaxf(v2, __shfl_xor(v2, off, 32));
        if (t == 0) s_b = v2;
    }
    __syncthreads();
    m = s_b;

    const float e = __expf(sc - m);
    float ssum = e;
#pragma unroll
    for (int off = 16; off > 0; off >>= 1) ssum += __shfl_xor(ssum, off, 32);
    if (lane == 0) s_red[wave] = ssum;
    __syncthreads();
    if (t < 16) {
        float v2 = s_red[t];
#pragma unroll
        for (int off = 8; off > 0; off >>= 1) v2 += __shfl_xor(v2, off, 32);
        if (t == 0) s_b = v2;
    }
    __syncthreads();
    const float sum = s_b;

    float prob = e / sum;
    prob = (1.0f - GAMMA) * prob + GAMMA / (float)NCAND;
    const float uu  = u[(size_t)b * NCAND + t];
    const float g   = -__logf(-__logf(uu + 1e-20f) + 1e-20f);
    float key = __logf(prob) + g;

    // ---- iterative top-32 argmax (descending order, lower-index tiebreak) ----
    for (int it = 0; it < KSAMP; ++it) {
        float bv = key;
        int   bi = t;
#pragma unroll
        for (int off = 16; off > 0; off >>= 1) {
            const float ov = __shfl_xor(bv, off, 32);
            const int   oi = __shfl_xor(bi, off, 32);
            if (ov > bv || (ov == bv && oi < bi)) { bv = ov; bi = oi; }
        }
        if (lane == 0) { s_red[wave] = bv; s_ridx[wave] = bi; }
        __syncthreads();
        if (t < 16) {
            float v2 = s_red[t];
            int   i2 = s_ridx[t];
#pragma unroll
            for (int off = 8; off > 0; off >>= 1) {
                const float ov = __shfl_xor(v2, off, 32);
                const int   oi = __shfl_xor(i2, off, 32);
                if (ov > v2 || (ov == v2 && oi < i2)) { v2 = ov; i2 = oi; }
            }
            if (t == 0) s_widx = i2;
        }
        __syncthreads();
        const int win = s_widx;
        if (t == 0) out[b * KSAMP + it] = win;
        if (t == win) key = -INFINITY;
        __syncthreads();
    }
}

// ---------------------------------------------------------------------------
extern "C" void kernel_launch(void* const* d_in, const int* in_sizes, int n_in,
                              void* d_out, int out_size, void* d_ws, size_t ws_size,
                              hipStream_t stream) {
    const float* tgt  = (const float*)d_in[0];   // [4096,128]
    const float* cand = (const float*)d_in[1];   // [4096,512,128]
    const float* Wq   = (const float*)d_in[2];   // [128,128]
    const float* bq   = (const float*)d_in[3];   // [128]
    const float* Wk   = (const float*)d_in[4];   // [128,128]
    // d_in[5] = bk: provably unused (softmax shift invariance)
    const float* u    = (const float*)d_in[6];   // [4096,512]

    float* Mmat = (float*)d_ws;                  // 128*128 floats
    float* cvec = Mmat + DIM * DIM;              // 128 floats
    float* V    = cvec + DIM;                    // 4096*128 floats (~2 MB)

    as_precompute_kernel<<<DIM, DIM, 0, stream>>>(Wq, bq, Wk, Mmat, cvec);

    dim3 g1(B_SZ / 16, DIM / 16);
    as_project_kernel<<<g1, 32, 0, stream>>>(tgt, Mmat, cvec, V);

    as_sampler_kernel<<<B_SZ, 512, 0, stream>>>(cand, V, u, (int*)d_out);
}